// MaskCoefficientTripletLoss_66211215835781
// MI455X (gfx1250) — compile-verified
//
#include <hip/hip_runtime.h>

// Problem constants (fixed by the reference setup_inputs()).
#define Bdim 4
#define Ndim 6400
#define Ddim 32
#define MARGIN 0.3f
#define EPSF 1e-12f

#define CHUNKS 4                       // candidate-dim split for parallelism
#define TPB (Ndim / 16)                // 400 candidate tiles per batch
#define TPC (TPB / CHUNKS)             // 100 tiles per chunk
#define ROWPAD 36                      // LDS floats per candidate row (32+4):
                                       // stride 36 -> 64 words/wave hit 64 banks

typedef float v2f __attribute__((ext_vector_type(2)));
typedef float v8f __attribute__((ext_vector_type(8)));
typedef int   v4i __attribute__((vector_size(16)));   // matches builtin param

#if __has_builtin(__builtin_amdgcn_global_load_async_to_lds_b128)
#define HAVE_ASYNC_LDS 1
#else
#define HAVE_ASYNC_LDS 0
#endif

__device__ __forceinline__ void lds_stage_b128(const float* g, float* l)
{
#if HAVE_ASYNC_LDS
    __builtin_amdgcn_global_load_async_to_lds_b128(
        (__attribute__((address_space(1))) v4i*)(void*)g,
        (__attribute__((address_space(3))) v4i*)(void*)l,
        0, 0);
#else
    *(float4*)l = *(const float4*)g;
#endif
}

__device__ __forceinline__ void wait_async_lds()
{
#if HAVE_ASYNC_LDS
#if __has_builtin(__builtin_amdgcn_s_wait_asynccnt)
    __builtin_amdgcn_s_wait_asynccnt(0);
#else
    asm volatile("s_wait_asynccnt 0x0" ::: "memory");
#endif
#endif
}

// Monotone float->u32 map for argmax-by-atomicMax.
__device__ __forceinline__ unsigned f2ord(float f)
{
    unsigned u = __float_as_uint(f);
    return (u & 0x80000000u) ? ~u : (u | 0x80000000u);
}

// ---------------------------------------------------------------------------
// Kernel 1: L2-normalize coefficients; build pk = fg ? gt : -1; zero the
// per-anchor argmax keys and the accumulators (every launch: graph replay).
// ---------------------------------------------------------------------------
__global__ void triplet_normalize(const float* __restrict__ pred,
                                  const int* __restrict__ gt,
                                  const int* __restrict__ fg,
                                  float* __restrict__ coeffs,
                                  int* __restrict__ pk,
                                  unsigned long long* __restrict__ packed,
                                  float* __restrict__ accum,
                                  int total)
{
    int i = blockIdx.x * blockDim.x + threadIdx.x;
    if (i < 2) accum[i] = 0.0f;
    if (i >= total) return;

    pk[i]     = fg[i] ? gt[i] : -1;
    packed[i] = 0ull;

    const float* p = pred + (size_t)i * Ddim;
    float v[Ddim];
    float s = 0.0f;
#pragma unroll
    for (int d = 0; d < Ddim; ++d) { v[d] = p[d]; s += v[d] * v[d]; }
    float scale = 1.0f / fmaxf(sqrtf(s), EPSF);

    float* c = coeffs + (size_t)i * Ddim;
#pragma unroll
    for (int d = 0; d < Ddim; ++d) c[d] = v[d] * scale;
}

// ---------------------------------------------------------------------------
// Kernel 2: hardest-negative mining. Block = 256 threads = 8 waves = 128
// anchors; candidate tiles double-buffered in LDS via async copies; 8
// v_wmma_f32_16x16x4_f32 per tile per wave; per-chunk argmax merged with a
// monotone u64 atomicMax (first-max tie-break preserved via ~idx).
// ---------------------------------------------------------------------------
__global__ void __launch_bounds__(256)
triplet_negmine(const float* __restrict__ coeffs,
                const int* __restrict__ pk,
                unsigned long long* __restrict__ packed)
{
    __shared__ float sbuf[2][16 * ROWPAD];   // double-buffered candidate tile
    __shared__ int   spkc[16 * TPC];         // pk for this chunk's candidates

    const int tid  = threadIdx.x;
    const int lane = tid & 31;
    const int wv   = tid >> 5;               // 0..7
    const int half = lane >> 4;              // 0 | 1
    const int l15  = lane & 15;

    int tmp = blockIdx.x;
    const int chunk = tmp % CHUNKS;  tmp /= CHUNKS;
    const int tb    = tmp % (Ndim / 128);
    const int b     = tmp / (Ndim / 128);

    const size_t bOff     = (size_t)b * Ndim;
    const int    aBase    = tb * 128 + wv * 16;
    const int    candBase = chunk * TPC * 16;

    // Anchor fragments: f32 16x4 A layout, K=32 in 8 chunks.
    // lane<16 -> k = 4c+0,4c+1 ; lane>=16 -> k = 4c+2,4c+3 (row = aBase+l15)
    v2f afrag[8];
    {
        const float* ap = coeffs + (bOff + aBase + l15) * Ddim + 2 * half;
#pragma unroll
        for (int c = 0; c < 8; ++c) afrag[c] = *(const v2f*)(ap + 4 * c);
    }

    int aGt[8];
#pragma unroll
    for (int r = 0; r < 8; ++r) aGt[r] = pk[bOff + aBase + r + 8 * half];
    // NB: aGt may be -1 for non-fg anchors; candidates have pv>=0, so
    // non-fg anchors just collect a harmless (unused) argmax.

    float bestSim[8];
    int   bestIdx[8];
#pragma unroll
    for (int r = 0; r < 8; ++r) { bestSim[r] = -3.0e38f; bestIdx[r] = -1; }

    const float* src0 = coeffs + (bOff + candBase) * Ddim;

    // Prologue: stage tile 0 (wave 0) while everyone stages the pk chunk.
    if (wv == 0) {
#pragma unroll
        for (int j = 0; j < 4; ++j) {
            int idx = j * 32 + lane, c = idx >> 3, ch = idx & 7;
            lds_stage_b128(src0 + c * Ddim + ch * 4, &sbuf[0][c * ROWPAD + ch * 4]);
        }
    }
    for (int i = tid; i < 16 * TPC; i += 256) spkc[i] = pk[bOff + candBase + i];
    if (wv == 0) wait_async_lds();
    __syncthreads();

    for (int t = 0; t < TPC; ++t) {
        // Stage tile t+1 into the other buffer while computing tile t.
        if (wv == 0 && (t + 1) < TPC) {
            const float* src = src0 + (size_t)(t + 1) * 16 * Ddim;
            float* dst = sbuf[(t + 1) & 1];
#pragma unroll
            for (int j = 0; j < 4; ++j) {
                int idx = j * 32 + lane, c = idx >> 3, ch = idx & 7;
                lds_stage_b128(src + c * Ddim + ch * 4, dst + c * ROWPAD + ch * 4);
            }
        }

        // Gather all 8 B fragments first (one DS wait), then 8 WMMAs.
        const float* buf = sbuf[t & 1];
        v2f bfrag[8];
#pragma unroll
        for (int c = 0; c < 8; ++c)
            bfrag[c] = *(const v2f*)(buf + l15 * ROWPAD + 4 * c + 2 * half);
        const int pv = spkc[t * 16 + l15];   // candidate fg?gt:-1

        v8f acc = {0.f, 0.f, 0.f, 0.f, 0.f, 0.f, 0.f, 0.f};
#pragma unroll
        for (int c = 0; c < 8; ++c)
            acc = __builtin_amdgcn_wmma_f32_16x16x4_f32(
                false, afrag[c], false, bfrag[c], (short)0, acc, false, false);

        if (pv >= 0) {
            const int m = candBase + t * 16 + l15;
#pragma unroll
            for (int r = 0; r < 8; ++r) {
                float s = acc[r];
                if (pv != aGt[r] && s > bestSim[r]) {  // strict >: first max
                    bestSim[r] = s;
                    bestIdx[r] = m;
                }
            }
        }

        if (wv == 0) wait_async_lds();
        __syncthreads();
    }

    // Reduce across the 16 lanes of each half (xor masks < 16 stay in-half),
    // then merge chunks via monotone u64 atomicMax.
#pragma unroll
    for (int r = 0; r < 8; ++r) {
        float v  = bestSim[r];
        int   ix = bestIdx[r];
#pragma unroll
        for (int off = 8; off >= 1; off >>= 1) {
            float ov = __shfl_xor(v, off, 32);
            int   oi = __shfl_xor(ix, off, 32);
            if (ov > v || (ov == v && oi < ix)) { v = ov; ix = oi; }
        }
        if (l15 == 0 && ix >= 0) {
            unsigned long long key =
                ((unsigned long long)f2ord(v) << 32) | (unsigned)(~(unsigned)ix);
            atomicMax(packed + bOff + aBase + r + 8 * half, key);
        }
    }
}

// ---------------------------------------------------------------------------
// Kernel 3: positive rank-select + triplet loss. Block = 256 threads = 8
// anchors of one batch; the batch's pk row is staged once in LDS.
// ---------------------------------------------------------------------------
__global__ void __launch_bounds__(256)
triplet_posloss(const float* __restrict__ coeffs,
                const float* __restrict__ pos_rand,
                const int* __restrict__ pk,
                const unsigned long long* __restrict__ packed,
                float* __restrict__ accum)
{
    __shared__ int spk[Ndim];                // 25.6 KB

    const int tid  = threadIdx.x;
    const int lane = tid & 31;
    const int wv   = tid >> 5;
    const int aIdx = blockIdx.x * 8 + wv;    // 8 anchors/block, same batch
    const int b    = aIdx / Ndim;
    const int n    = aIdx % Ndim;
    const size_t bOff = (size_t)b * Ndim;

    for (int i = tid; i < Ndim / 4; i += 256)
        ((int4*)spk)[i] = ((const int4*)(pk + bOff))[i];
    __syncthreads();

    const int aPk = spk[n];
    if (aPk < 0) return;                     // anchor not fg (wave-uniform)
    const int aGt = aPk;

    // Pass 1: count positives, detect any negative.
    int posCnt = 0, hasNeg = 0;
    for (int m = lane; m < Ndim; m += 32) {
        int pv = spk[m];
        if (pv >= 0) {
            if (pv == aGt) { if (m != n) posCnt++; }
            else hasNeg = 1;
        }
    }
#pragma unroll
    for (int off = 16; off >= 1; off >>= 1) {
        posCnt += __shfl_xor(posCnt, off, 32);
        hasNeg |= __shfl_xor(hasNeg, off, 32);
    }
    const int numPos = posCnt;
    if (numPos < 1 || !hasNeg) return;       // wave-uniform

    // k-th positive in anchor order, k = clip(floor(u*numPos), 0, numPos-1)
    float u = pos_rand[bOff + n];
    int k = (int)floorf(u * (float)numPos);
    if (k < 0) k = 0;
    if (k > numPos - 1) k = numPos - 1;
    const int target = k + 1;

    // Pass 2: ordered rank-select via wave32 ballot over 32-wide chunks.
    int posM = 0, running = 0;
    for (int c = 0; c * 32 < Ndim; ++c) {
        int m = c * 32 + lane;
        int pv = spk[m];
        int isPos = (pv == aGt) && (m != n);
        unsigned mask = (unsigned)__ballot(isPos);
        int cnt = __popc(mask);
        if (running + cnt >= target) {
            int rank = __popc(mask & ((1u << lane) - 1u)) + 1;
            int sel = isPos && (running + rank == target);
            unsigned smask = (unsigned)__ballot(sel);
            posM = __shfl(m, __ffs(smask) - 1, 32);
            break;
        }
        running += cnt;
    }

    // Decode hardest negative from the packed argmax key.
    const unsigned long long key = packed[bOff + n];
    const int negM = (int)(~(unsigned)(key & 0xffffffffull));

    // Distances: D == 32 == wave width, one element per lane.
    float ca = coeffs[(bOff + n)    * Ddim + lane];
    float cp = coeffs[(bOff + posM) * Ddim + lane];
    float cn = coeffs[(bOff + negM) * Ddim + lane];
    float dp = ca - cp, dn = ca - cn;
    float sp = dp * dp, sn = dn * dn;
#pragma unroll
    for (int off = 16; off >= 1; off >>= 1) {
        sp += __shfl_xor(sp, off, 32);
        sn += __shfl_xor(sn, off, 32);
    }
    if (lane == 0) {
        float per = fmaxf(sqrtf(sp + EPSF) - sqrtf(sn + EPSF) + MARGIN, 0.0f);
        atomicAdd(&accum[0], per);
        atomicAdd(&accum[1], 1.0f);
    }
}

// ---------------------------------------------------------------------------
// Kernel 4: finalize scalar output.
// ---------------------------------------------------------------------------
__global__ void triplet_finalize(const float* __restrict__ accum,
                                 float* __restrict__ out)
{
    float total = accum[0];
    float cnt   = accum[1];
    out[0] = (cnt > 0.0f) ? (total / cnt) : 0.0f;
}

// ---------------------------------------------------------------------------
extern "C" void kernel_launch(void* const* d_in, const int* in_sizes, int n_in,
                              void* d_out, int out_size, void* d_ws, size_t ws_size,
                              hipStream_t stream)
{
    (void)in_sizes; (void)n_in; (void)out_size; (void)ws_size;

    const float* pred  = (const float*)d_in[0];   // [B,N,32] f32
    const float* prand = (const float*)d_in[1];   // [B,N]    f32
    const int*   gt    = (const int*)d_in[2];     // [B,N]    int
    const int*   fg    = (const int*)d_in[3];     // [B,N]    int (bool)
    float*       out   = (float*)d_out;           // scalar

    char* ws = (char*)d_ws;
    float* coeffs = (float*)ws;                ws += (size_t)Bdim * Ndim * Ddim * sizeof(float);
    int*   pk     = (int*)ws;                  ws += (size_t)Bdim * Ndim * sizeof(int);
    unsigned long long* packed =
        (unsigned long long*)ws;               ws += (size_t)Bdim * Ndim * sizeof(unsigned long long);
    float* accum  = (float*)ws;

    const int total = Bdim * Ndim;

    triplet_normalize<<<(total + 255) / 256, 256, 0, stream>>>(
        pred, gt, fg, coeffs, pk, packed, accum, total);

    triplet_negmine<<<Bdim * (Ndim / 128) * CHUNKS, 256, 0, stream>>>(
        coeffs, pk, packed);

    triplet_posloss<<<total / 8, 256, 0, stream>>>(
        coeffs, prand, pk, packed, accum);

    triplet_finalize<<<1, 1, 0, stream>>>(accum, out);
}